// MGCN_78400333021783
// MI455X (gfx1250) — compile-verified
//
#include <hip/hip_runtime.h>
#include <hip/hip_bf16.h>

#define NN 10000
#define BB 64
#define DD 64
#define UU 64

typedef __attribute__((ext_vector_type(2))) float v2f;
typedef __attribute__((ext_vector_type(8))) float v8f;

// ---------------------------------------------------------------------------
// Kernel 1: per-node GEMMs  C_mx[b,u] = sum_d x[b,n,d] * K_mx[d,u]
//   mx=0 -> d_out (+bias), mx=1 -> ws[0..], mx=2 -> ws[N*4096..]
// One block per node n.  128 threads = 4 waves; wave w owns b-strip [16w,16w+16).
// Uses V_WMMA_F32_16X16X4_F32 (fp32, M=N=16, K=4), K-loop of 16 steps.
// ---------------------------------------------------------------------------
__global__ __launch_bounds__(128) void MGCN_fused_gemm(
    const float* __restrict__ x,      // [B, N, D]
    const float* __restrict__ kern,   // [3*D, U]
    const float* __restrict__ bias,   // [U]
    float* __restrict__ out,          // [B, N, U]
    float* __restrict__ ws)           // [2, N, B*U]
{
    __shared__ float sk[3 * DD * UU];   // 48KB: whole weight matrix in LDS

    const int n   = blockIdx.x;
    const int tid = threadIdx.x;

    for (int i = tid; i < 3 * DD * UU; i += 128) sk[i] = kern[i];
    __syncthreads();

    const int wave = tid >> 5;
    const int lane = tid & 31;
    const int hh   = lane >> 4;     // lane half: 0 or 1
    const int l15  = lane & 15;
    const int b0   = wave * 16;     // b-strip base

    // A row for this lane: A[b,d] = x[(b*N + n)*D + d], row M = b0 + l15
    const float* arow = x + ((size_t)(b0 + l15) * NN + (size_t)n) * DD;

    v8f acc[12];                    // [mx][utile] -> acc[mx*4 + t]
#pragma unroll
    for (int i = 0; i < 12; ++i) acc[i] = (v8f){0.f,0.f,0.f,0.f,0.f,0.f,0.f,0.f};

    for (int k = 0; k < DD; k += 4) {
        // A operand: lanes 0-15 hold K=k,k+1 ; lanes 16-31 hold K=k+2,k+3
        v2f a = *(const v2f*)(arow + k + 2 * hh);
        const int r = k + 2 * hh;   // B operand rows r (x) and r+1 (y)
#pragma unroll
        for (int m = 0; m < 3; ++m) {
            const float* km = sk + m * DD * UU;
#pragma unroll
            for (int t = 0; t < 4; ++t) {
                v2f bb;
                bb.x = km[(r    ) * UU + t * 16 + l15];
                bb.y = km[(r + 1) * UU + t * 16 + l15];
                acc[m * 4 + t] = __builtin_amdgcn_wmma_f32_16x16x4_f32(
                    false, a, false, bb, (short)0, acc[m * 4 + t], false, false);
            }
        }
    }

    // C/D layout: VGPR j -> row M = b0 + 8*hh + j, col = utile*16 + l15
    const int Mbase = b0 + 8 * hh;
    float bv[4];
#pragma unroll
    for (int t = 0; t < 4; ++t) bv[t] = bias[t * 16 + l15];

    float* y1 = ws + (size_t)n * (BB * UU);
    float* y2 = ws + (size_t)NN * (BB * UU) + (size_t)n * (BB * UU);

#pragma unroll
    for (int j = 0; j < 8; ++j) {
        const int brow = Mbase + j;
        const size_t obase = ((size_t)brow * NN + (size_t)n) * UU;
#pragma unroll
        for (int t = 0; t < 4; ++t) {
            out[obase + t * 16 + l15]      = acc[0 * 4 + t][j] + bv[t];
            y1[brow * UU + t * 16 + l15]   = acc[1 * 4 + t][j];
            y2[brow * UU + t * 16 + l15]   = acc[2 * 4 + t][j];
        }
    }
}

// ---------------------------------------------------------------------------
// Kernel 2: COO scatter-accumulate.  For edge e:
//   out[b, rows[e], u] += vals[e] * Y[cols[e], b*64+u]   for all 4096 features
// One block per nnz; 256 threads x 4 float4 iterations; fp32 HW atomics.
// ---------------------------------------------------------------------------
__global__ __launch_bounds__(256) void MGCN_spmm_acc(
    const int*   __restrict__ rows,
    const int*   __restrict__ cols,
    const float* __restrict__ vals,
    const float* __restrict__ Y,      // [N, 4096] = [N, b, u]
    float* __restrict__ out,          // [B, N, U]
    int nnz)
{
    const int e = blockIdx.x;
    if (e >= nnz) return;

    const int   r = rows[e];
    const int   c = cols[e];
    const float v = vals[e];

    const float4* __restrict__ src = (const float4*)(Y + (size_t)c * (BB * UU));
    float* dstrow = out + (size_t)r * UU;

#pragma unroll
    for (int it = 0; it < 4; ++it) {
        const int i = threadIdx.x + it * 256;   // float4 index, f = 4*i
        float4 t = src[i];
        const int b = i >> 4;                   // f / 64
        const int u = (i & 15) * 4;             // f % 64
        float* dst = dstrow + (size_t)b * (NN * UU) + u;
        unsafeAtomicAdd(dst + 0, v * t.x);
        unsafeAtomicAdd(dst + 1, v * t.y);
        unsafeAtomicAdd(dst + 2, v * t.z);
        unsafeAtomicAdd(dst + 3, v * t.w);
    }
}

extern "C" void kernel_launch(void* const* d_in, const int* in_sizes, int n_in,
                              void* d_out, int out_size, void* d_ws, size_t ws_size,
                              hipStream_t stream) {
    const float* x    = (const float*)d_in[0];
    const int*   r0   = (const int*)  d_in[1];
    const int*   c0   = (const int*)  d_in[2];
    const float* v0   = (const float*)d_in[3];
    const int*   r1   = (const int*)  d_in[4];
    const int*   c1   = (const int*)  d_in[5];
    const float* v1   = (const float*)d_in[6];
    const float* kern = (const float*)d_in[7];
    const float* bias = (const float*)d_in[8];

    float* out = (float*)d_out;
    float* ws  = (float*)d_ws;   // needs 2 * N * 4096 floats (~328 MB)

    // Pass 1: per-node GEMMs; writes mx=0 (+bias) into out, Y1/Y2 into ws.
    MGCN_fused_gemm<<<NN, 128, 0, stream>>>(x, kern, bias, out, ws);

    // Pass 2: two sequential SpMM scatter-adds (separate launches so each
    // 164MB Y buffer can stay resident in the 192MB L2 during its pass).
    const int nnz0 = in_sizes[1];
    const int nnz1 = in_sizes[4];
    MGCN_spmm_acc<<<nnz0, 256, 0, stream>>>(r0, c0, v0, ws, out, nnz0);
    MGCN_spmm_acc<<<nnz1, 256, 0, stream>>>(r1, c1, v1,
                                            ws + (size_t)NN * (BB * UU), out, nnz1);
}